// PACNet_28630251995361
// MI455X (gfx1250) — compile-verified
//
#include <hip/hip_runtime.h>
#include <hip/hip_bf16.h>

typedef __attribute__((ext_vector_type(2))) float v2f;
typedef __attribute__((ext_vector_type(8))) float v8f;

#define NEG_SLOPE 0.2f
#define ENC_NEG_INF 0x007FFFFFu  // encodeOrd(-inf)

// ---- order-preserving float<->uint encoding for atomicMax on floats ----
__device__ __forceinline__ unsigned encodeOrd(float f) {
  unsigned u = __float_as_uint(f);
  return (u & 0x80000000u) ? ~u : (u | 0x80000000u);
}
__device__ __forceinline__ float decodeOrd(unsigned u) {
  unsigned b = (u & 0x80000000u) ? (u & 0x7FFFFFFFu) : ~u;
  return __uint_as_float(b);
}
__device__ __forceinline__ void atomicMaxOrd(unsigned* p, float f) {
  atomicMax(p, encodeOrd(f));
}
__device__ __forceinline__ float finiteOr0(float f) {
  return ((__float_as_uint(f) & 0x7F800000u) != 0x7F800000u) ? f : 0.0f;
}

// ---------------- fills ----------------
__global__ void fill_f32_k(float* p, float v, int n) {
  int i = blockIdx.x * blockDim.x + threadIdx.x;
  if (i < n) p[i] = v;
}
__global__ void fill_u32_k(unsigned* p, unsigned v, int n) {
  int i = blockIdx.x * blockDim.x + threadIdx.x;
  if (i < n) p[i] = v;
}
__global__ void init_out_k(float* out, const float* __restrict__ b2, int total) {
  int i = blockIdx.x * blockDim.x + threadIdx.x;
  if (i < total) out[i] = b2[i & 31];
}

// ---------------- layer 1 GEMM: h[n,64] = x[n,300] @ W1[300,64] (fp32 WMMA) ----------------
__global__ __launch_bounds__(128) void gemm1_wmma_k(
    const float* __restrict__ x, const float* __restrict__ W1,
    float* __restrict__ h, int n) {
  const int lane = threadIdx.x & 31;
  const int wave = threadIdx.x >> 5;
  const int rowTiles = (n + 15) >> 4;
  const int rt = blockIdx.x * 4 + wave;
  if (rt >= rowTiles) return;              // wave-uniform exit; EXEC all-1 inside
  const int m0 = rt << 4;
  const int halfSel = lane >> 4;           // 0: lanes 0-15 (K pair 0,1), 1: lanes 16-31 (K pair 2,3)
  const int l15 = lane & 15;
  const int arow = min(m0 + l15, n - 1);   // clamp so loads stay unpredicated
  const int koff = halfSel * 2;
  const float* xrow = x + (size_t)arow * 300;

  v8f c0 = {}, c1 = {}, c2 = {}, c3 = {};
  for (int k = 0; k < 300; k += 4) {
    // A 16x4 fp32: lane-half selects K pair, v2f -> 2 consecutive VGPRs
    v2f a = *(const v2f*)(xrow + k + koff);
    const float* wk = W1 + (size_t)(k + koff) * 64 + l15;  // B 4x16 tiles, row stride 64
    v2f b0, b1, b2, b3;
    b0.x = wk[0];  b0.y = wk[64];
    b1.x = wk[16]; b1.y = wk[80];
    b2.x = wk[32]; b2.y = wk[96];
    b3.x = wk[48]; b3.y = wk[112];
    c0 = __builtin_amdgcn_wmma_f32_16x16x4_f32(false, a, false, b0, (short)0, c0, false, false);
    c1 = __builtin_amdgcn_wmma_f32_16x16x4_f32(false, a, false, b1, (short)0, c1, false, false);
    c2 = __builtin_amdgcn_wmma_f32_16x16x4_f32(false, a, false, b2, (short)0, c2, false, false);
    c3 = __builtin_amdgcn_wmma_f32_16x16x4_f32(false, a, false, b3, (short)0, c3, false, false);
  }
#pragma unroll
  for (int r = 0; r < 8; ++r) {
    int row = m0 + halfSel * 8 + r;        // C/D layout: VGPR r holds M=r (lo half) / M=r+8 (hi half)
    if (row < n) {
      float* hr = h + (size_t)row * 64 + l15;
      hr[0]  = c0[r];
      hr[16] = c1[r];
      hr[32] = c2[r];
      hr[48] = c3[r];
    }
  }
}

// ---------------- layer 2 GEMM: h2[n,32] = relu(out1+b1)[n,64] @ W2[64,32] ----------------
__global__ __launch_bounds__(128) void gemm2_wmma_k(
    const float* __restrict__ o1, const float* __restrict__ bias1,
    const float* __restrict__ W2, float* __restrict__ h2, int n) {
  const int lane = threadIdx.x & 31;
  const int wave = threadIdx.x >> 5;
  const int rowTiles = (n + 15) >> 4;
  const int rt = blockIdx.x * 4 + wave;
  if (rt >= rowTiles) return;
  const int m0 = rt << 4;
  const int halfSel = lane >> 4;
  const int l15 = lane & 15;
  const int arow = min(m0 + l15, n - 1);
  const int koff = halfSel * 2;
  const float* xrow = o1 + (size_t)arow * 64;

  v8f c0 = {}, c1 = {};
  for (int k = 0; k < 64; k += 4) {
    int ka = k + koff;
    v2f a;                                  // fused bias + relu on the fly
    a.x = fmaxf(xrow[ka]     + bias1[ka],     0.0f);
    a.y = fmaxf(xrow[ka + 1] + bias1[ka + 1], 0.0f);
    const float* wk = W2 + (size_t)ka * 32 + l15;
    v2f b0, b1;
    b0.x = wk[0];  b0.y = wk[32];
    b1.x = wk[16]; b1.y = wk[48];
    c0 = __builtin_amdgcn_wmma_f32_16x16x4_f32(false, a, false, b0, (short)0, c0, false, false);
    c1 = __builtin_amdgcn_wmma_f32_16x16x4_f32(false, a, false, b1, (short)0, c1, false, false);
  }
#pragma unroll
  for (int r = 0; r < 8; ++r) {
    int row = m0 + halfSel * 8 + r;
    if (row < n) {
      float* hr = h2 + (size_t)row * 32 + l15;
      hr[0]  = c0[r];
      hr[16] = c1[r];
    }
  }
}

// ---------------- layer 1 attention: logits + segment max ----------------
__global__ void att1_logits_k(const long long* __restrict__ esrc,
                              const long long* __restrict__ edst,
                              long long eraw, long long etot,
                              const float* __restrict__ h,
                              const float* __restrict__ a1,
                              float* __restrict__ elog,
                              unsigned* __restrict__ mx) {
  long long t = (long long)blockIdx.x * blockDim.x + threadIdx.x;
  if (t >= etot * 4) return;
  long long e = t >> 2;
  int hd = (int)(t & 3);
  long long s, d;
  if (e < eraw) { s = esrc[e]; d = edst[e]; } else { s = e - eraw; d = s; }
  const float* hdv = h + (size_t)d * 64 + hd * 16;
  const float* hsv = h + (size_t)s * 64 + hd * 16;
  const float* ai = a1 + hd * 32;
  const float* aj = ai + 16;
  float acc = 0.0f;
#pragma unroll
  for (int q = 0; q < 16; ++q) acc += hdv[q] * ai[q] + hsv[q] * aj[q];
  float l = (acc > 0.0f) ? acc : NEG_SLOPE * acc;
  elog[t] = l;
  atomicMaxOrd(&mx[(size_t)d * 4 + hd], l);
}

// ---------------- layer 1: exp(l - m) and segment sum ----------------
__global__ void att1_expsum_k(const long long* __restrict__ edst,
                              long long eraw, long long etot,
                              const unsigned* __restrict__ mx,
                              float* __restrict__ elog,
                              float* __restrict__ sum) {
  long long t = (long long)blockIdx.x * blockDim.x + threadIdx.x;
  if (t >= etot * 4) return;
  long long e = t >> 2;
  int hd = (int)(t & 3);
  long long d = (e < eraw) ? edst[e] : (e - eraw);
  float m = finiteOr0(decodeOrd(mx[(size_t)d * 4 + hd]));
  float ex = __expf(elog[t] - m);
  elog[t] = ex;
  atomicAdd(&sum[(size_t)d * 4 + hd], ex);
}

// ---------------- layer 1: out1 += alpha * h[src] ----------------
__global__ void att1_aggr_k(const long long* __restrict__ esrc,
                            const long long* __restrict__ edst,
                            long long eraw, long long etot,
                            const float* __restrict__ elog,
                            const float* __restrict__ sum,
                            const float* __restrict__ h,
                            float* __restrict__ o1) {
  long long t = (long long)blockIdx.x * blockDim.x + threadIdx.x;
  if (t >= etot * 4) return;
  long long e = t >> 2;
  int hd = (int)(t & 3);
  long long s, d;
  if (e < eraw) { s = esrc[e]; d = edst[e]; } else { s = e - eraw; d = s; }
  float alpha = elog[t] / (sum[(size_t)d * 4 + hd] + 1e-16f);
  const float* hsv = h + (size_t)s * 64 + hd * 16;
  float* ov = o1 + (size_t)d * 64 + hd * 16;
#pragma unroll
  for (int q = 0; q < 16; ++q) atomicAdd(&ov[q], alpha * hsv[q]);
}

// ---------------- layer 2 attention: logits + segment max ----------------
__global__ void att2_logits_k(const long long* __restrict__ pi,
                              const long long* __restrict__ pj,
                              const long long* __restrict__ pk,
                              long long m2,
                              const float* __restrict__ h2,
                              const float* __restrict__ a2,
                              float* __restrict__ elog,
                              unsigned* __restrict__ mx) {
  long long t = (long long)blockIdx.x * blockDim.x + threadIdx.x;
  if (t >= m2 * 4) return;
  long long e = t >> 2;
  int hd = (int)(t & 3);
  long long i = pi[e], j = pj[e], kk = pk[e];
  const float* hi = h2 + (size_t)i * 32 + hd * 8;
  const float* hj = h2 + (size_t)j * 32 + hd * 8;
  const float* hk = h2 + (size_t)kk * 32 + hd * 8;
  const float* ai = a2 + hd * 24;
  const float* aj = ai + 8;
  const float* ak = ai + 16;
  float acc = 0.0f;
#pragma unroll
  for (int q = 0; q < 8; ++q) acc += hi[q] * ai[q] + hj[q] * aj[q] + hk[q] * ak[q];
  float l = (acc > 0.0f) ? acc : NEG_SLOPE * acc;
  elog[t] = l;
  atomicMaxOrd(&mx[(size_t)i * 4 + hd], l);
}

__global__ void att2_expsum_k(const long long* __restrict__ pi, long long m2,
                              const unsigned* __restrict__ mx,
                              float* __restrict__ elog,
                              float* __restrict__ sum) {
  long long t = (long long)blockIdx.x * blockDim.x + threadIdx.x;
  if (t >= m2 * 4) return;
  long long e = t >> 2;
  int hd = (int)(t & 3);
  long long i = pi[e];
  float m = finiteOr0(decodeOrd(mx[(size_t)i * 4 + hd]));
  float ex = __expf(elog[t] - m);
  elog[t] = ex;
  atomicAdd(&sum[(size_t)i * 4 + hd], ex);
}

__global__ void att2_aggr_k(const long long* __restrict__ pi,
                            const long long* __restrict__ pk,
                            long long m2,
                            const float* __restrict__ elog,
                            const float* __restrict__ sum,
                            const float* __restrict__ h2,
                            float* __restrict__ out) {
  long long t = (long long)blockIdx.x * blockDim.x + threadIdx.x;
  if (t >= m2 * 4) return;
  long long e = t >> 2;
  int hd = (int)(t & 3);
  long long i = pi[e], kk = pk[e];
  float alpha = elog[t] / (sum[(size_t)i * 4 + hd] + 1e-16f);
  const float* hk = h2 + (size_t)kk * 32 + hd * 8;
  float* ov = out + (size_t)i * 32 + hd * 8;
#pragma unroll
  for (int q = 0; q < 8; ++q) atomicAdd(&ov[q], alpha * hk[q]);
}

// ---------------- host side ----------------
static inline unsigned gridFor(long long total, int block) {
  return (unsigned)((total + block - 1) / block);
}

extern "C" void kernel_launch(void* const* d_in, const int* in_sizes, int n_in,
                              void* d_out, int out_size, void* d_ws, size_t ws_size,
                              hipStream_t stream) {
  const float*      x   = (const float*)d_in[0];
  const long long*  ei  = (const long long*)d_in[1];
  const long long*  soi = (const long long*)d_in[2];
  const float*      W1  = (const float*)d_in[3];
  const float*      a1  = (const float*)d_in[4];
  const float*      b1  = (const float*)d_in[5];
  const float*      W2  = (const float*)d_in[6];
  const float*      a2  = (const float*)d_in[7];
  const float*      b2  = (const float*)d_in[8];
  float*            out = (float*)d_out;

  const int n = in_sizes[0] / 300;
  const long long eraw = in_sizes[1] / 2;
  const long long m2   = in_sizes[2] / 3;
  const long long etot = eraw + n;

  const long long* esrc = ei;
  const long long* edst = ei + eraw;
  const long long* pi = soi;
  const long long* pj = soi + m2;
  const long long* pk = soi + 2 * m2;

  // workspace layout (floats)
  float* ws = (float*)d_ws;
  float*    h    = ws;                              // n*64
  float*    o1   = h + (size_t)n * 64;              // n*64
  float*    h2   = o1 + (size_t)n * 64;             // n*32
  unsigned* mx1  = (unsigned*)(h2 + (size_t)n * 32);// n*4
  float*    sm1  = (float*)(mx1 + (size_t)n * 4);   // n*4
  unsigned* mx2  = (unsigned*)(sm1 + (size_t)n * 4);// n*4
  float*    sm2  = (float*)(mx2 + (size_t)n * 4);   // n*4
  float*    elog = sm2 + (size_t)n * 4;             // max(etot*4, m2*4)

  const int B = 256;

  // init
  fill_f32_k<<<gridFor((long long)n * 64, B), B, 0, stream>>>(o1, 0.0f, n * 64);
  fill_u32_k<<<gridFor((long long)n * 4, B), B, 0, stream>>>(mx1, ENC_NEG_INF, n * 4);
  fill_f32_k<<<gridFor((long long)n * 4, B), B, 0, stream>>>(sm1, 0.0f, n * 4);
  fill_u32_k<<<gridFor((long long)n * 4, B), B, 0, stream>>>(mx2, ENC_NEG_INF, n * 4);
  fill_f32_k<<<gridFor((long long)n * 4, B), B, 0, stream>>>(sm2, 0.0f, n * 4);
  init_out_k<<<gridFor((long long)n * 32, B), B, 0, stream>>>(out, b2, n * 32);

  // layer 1
  const int rowTiles = (n + 15) / 16;
  const unsigned gGemm = (unsigned)((rowTiles + 3) / 4);
  gemm1_wmma_k<<<gGemm, 128, 0, stream>>>(x, W1, h, n);
  att1_logits_k<<<gridFor(etot * 4, B), B, 0, stream>>>(esrc, edst, eraw, etot, h, a1, elog, mx1);
  att1_expsum_k<<<gridFor(etot * 4, B), B, 0, stream>>>(edst, eraw, etot, mx1, elog, sm1);
  att1_aggr_k<<<gridFor(etot * 4, B), B, 0, stream>>>(esrc, edst, eraw, etot, elog, sm1, h, o1);

  // layer 2
  gemm2_wmma_k<<<gGemm, 128, 0, stream>>>(o1, b1, W2, h2, n);
  att2_logits_k<<<gridFor(m2 * 4, B), B, 0, stream>>>(pi, pj, pk, m2, h2, a2, elog, mx2);
  att2_expsum_k<<<gridFor(m2 * 4, B), B, 0, stream>>>(pi, m2, mx2, elog, sm2);
  att2_aggr_k<<<gridFor(m2 * 4, B), B, 0, stream>>>(pi, pk, m2, elog, sm2, h2, out);
}